// Encoder_78718160601171
// MI455X (gfx1250) — compile-verified
//
#include <hip/hip_runtime.h>

// Row chunks as native 128-bit vectors so loads/stores lower to
// GLOBAL_LOAD_B128 / GLOBAL_STORE_B128 and __builtin_nontemporal_store works.
typedef float v4f __attribute__((ext_vector_type(4)));
typedef int   v4i __attribute__((ext_vector_type(4)));

#define PG_AS1 __attribute__((address_space(1)))
#define PG_AS3 __attribute__((address_space(3)))

// Async global<->LDS DMA path (ASYNCcnt) — gfx1250-specific. Guarded so the
// kernel always compiles even if a toolchain lacks the builtins.
#if defined(__HIP_DEVICE_COMPILE__) && defined(__gfx1250__)
#  if __has_builtin(__builtin_amdgcn_global_load_async_to_lds_b128) && \
      __has_builtin(__builtin_amdgcn_global_store_async_from_lds_b128) && \
      __has_builtin(__builtin_amdgcn_s_wait_asynccnt)
#    define PG_ASYNC 1
#  endif
#endif
#ifndef PG_ASYNC
#  define PG_ASYNC 0
#endif

// ---------------------------------------------------------------------------
// Specialized kernel for K=128 (KC==32): one wave32 per index per iteration.
//  - pm row:  pure copy -> async DMA global->LDS->global (never touches VGPRs)
//  - pv row:  plain b128 load -> v_exp_f32 -> non-temporal b128 store
// The pv VALU work overlaps the async DMA latency; ASYNCcnt<=1 before each
// drain store guarantees the staged row has landed in LDS (loads complete
// in order among loads); 4 LDS buffers per wave make the slot-reuse distance
// safe vs unordered async LDS accesses.
// ---------------------------------------------------------------------------
__global__ __launch_bounds__(256) void prior_gather_rows_k128(
    const int* __restrict__ indices,
    const v4f* __restrict__ Wm,   // [N, 32] float4
    const v4f* __restrict__ Wv,   // [N, 32] float4
    v4f* __restrict__ pm,         // [B, 32] float4
    v4f* __restrict__ pv,         // [B, 32] float4
    int B, int nwaves)
{
    const int lane  = (int)(threadIdx.x & 31u);
    const int wslot = (int)(threadIdx.x >> 5);   // wave within block (0..7)
    (void)wslot;
    const int wid = __builtin_amdgcn_readfirstlane(
        (int)((blockIdx.x * blockDim.x + threadIdx.x) >> 5));

#if PG_ASYNC
    __shared__ v4f stage[8][4][32];              // 16 KB: 4 buffers per wave

    int it = 0;
    int prev_b = -1;
    for (int b = wid; b < B; b += nwaves, ++it) {
        const int idx  = indices[b];             // wave-uniform -> scalar load
        const int slot = it & 3;

        // (1) Kick the pm-row DMA: 32 lanes x b128 = full 512B row, 1 ASYNCcnt.
        __builtin_amdgcn_global_load_async_to_lds_b128(
            (PG_AS1 v4i*)(Wm + (size_t)idx * 32 + lane),
            (PG_AS3 v4i*)&stage[wslot][slot][lane], 0, 0);

        // Prefetch next iteration's rows to hide random-row HBM latency.
        const int bn = b + nwaves;
        if (bn < B) {
            const int idxn = indices[bn];
            __builtin_prefetch(Wv + (size_t)idxn * 32 + lane, 0, 3);
            __builtin_prefetch(Wm + (size_t)idxn * 32 + lane, 0, 3);
        }

        // (2) pv row: plain load + exp + NT store (overlaps the DMA).
        v4f v = Wv[(size_t)idx * 32 + lane];
        v4f e;
        e.x = __expf(2.0f * v.x);
        e.y = __expf(2.0f * v.y);
        e.z = __expf(2.0f * v.z);
        e.w = __expf(2.0f * v.w);
        __builtin_nontemporal_store(e, &pv[(size_t)b * 32 + lane]);

        // (3) Drain the previous staged row LDS -> pm (async store).
        if (prev_b >= 0) {
            __builtin_amdgcn_s_wait_asynccnt(1);  // prev load definitely done
            const int pslot = (it - 1) & 3;
            __builtin_amdgcn_global_store_async_from_lds_b128(
                (PG_AS1 v4i*)(pm + (size_t)prev_b * 32 + lane),
                (PG_AS3 v4i*)&stage[wslot][pslot][lane], 0, 0);
        }
        prev_b = b;
    }
    if (prev_b >= 0) {                            // tail drain
        __builtin_amdgcn_s_wait_asynccnt(0);
        const int pslot = (it - 1) & 3;
        __builtin_amdgcn_global_store_async_from_lds_b128(
            (PG_AS1 v4i*)(pm + (size_t)prev_b * 32 + lane),
            (PG_AS3 v4i*)&stage[wslot][pslot][lane], 0, 0);
    }
    // s_endpgm performs an implicit wait-idle, covering in-flight async stores.
#else
    // Fallback: plain VMEM path (also what the host pass parses).
    for (int b = wid; b < B; b += nwaves) {
        const int idx = indices[b];
        const int bn = b + nwaves;
        if (bn < B) {
            const int idxn = indices[bn];
            __builtin_prefetch(Wm + (size_t)idxn * 32 + lane, 0, 3);
            __builtin_prefetch(Wv + (size_t)idxn * 32 + lane, 0, 3);
        }
        v4f m = Wm[(size_t)idx * 32 + lane];
        v4f v = Wv[(size_t)idx * 32 + lane];
        v4f e;
        e.x = __expf(2.0f * v.x);
        e.y = __expf(2.0f * v.y);
        e.z = __expf(2.0f * v.z);
        e.w = __expf(2.0f * v.w);
        __builtin_nontemporal_store(m, &pm[(size_t)b * 32 + lane]);
        __builtin_nontemporal_store(e, &pv[(size_t)b * 32 + lane]);
    }
#endif
}

// ---------------------------------------------------------------------------
// Generic-K fallback (any K % 4 == 0), plain VMEM path.
// ---------------------------------------------------------------------------
__global__ __launch_bounds__(256) void prior_gather_rows_generic(
    const int* __restrict__ indices,
    const v4f* __restrict__ Wm, const v4f* __restrict__ Wv,
    v4f* __restrict__ pm, v4f* __restrict__ pv,
    int B, int KC, int nwaves)
{
    const int lane = (int)(threadIdx.x & 31u);
    const int wid = __builtin_amdgcn_readfirstlane(
        (int)((blockIdx.x * blockDim.x + threadIdx.x) >> 5));

    for (int b = wid; b < B; b += nwaves) {
        const int idx = indices[b];
        const int bn = b + nwaves;
        if (bn < B) {
            const int idxn = indices[bn];
            const v4f* nm = Wm + (size_t)idxn * (size_t)KC;
            const v4f* nv = Wv + (size_t)idxn * (size_t)KC;
            for (int c = lane; c < KC; c += 32) {
                __builtin_prefetch(&nm[c], 0, 3);
                __builtin_prefetch(&nv[c], 0, 3);
            }
        }
        const v4f* rm = Wm + (size_t)idx * (size_t)KC;
        const v4f* rv = Wv + (size_t)idx * (size_t)KC;
        v4f* om = pm + (size_t)b * (size_t)KC;
        v4f* ov = pv + (size_t)b * (size_t)KC;
        for (int c = lane; c < KC; c += 32) {
            v4f m = rm[c];
            v4f v = rv[c];
            v4f e;
            e.x = __expf(2.0f * v.x);
            e.y = __expf(2.0f * v.y);
            e.z = __expf(2.0f * v.z);
            e.w = __expf(2.0f * v.w);
            __builtin_nontemporal_store(m, &om[c]);
            __builtin_nontemporal_store(e, &ov[c]);
        }
    }
}

// [B,1] columns: coalesced index loads / output stores; the 4B gathers are
// inherently scattered but total only ~1.6MB of the 412MB.
__global__ __launch_bounds__(256) void prior_gather_het(
    const int* __restrict__ indices,
    const float* __restrict__ Wmh,
    const float* __restrict__ Wvh,
    float* __restrict__ hm,
    float* __restrict__ hv,
    int B)
{
    const int t = (int)(blockIdx.x * blockDim.x + threadIdx.x);
    if (t < B) {
        const int idx = indices[t];
        const float m = Wmh[idx];
        const float e = __expf(2.0f * Wvh[idx]);
        __builtin_nontemporal_store(m, &hm[t]);
        __builtin_nontemporal_store(e, &hv[t]);
    }
}

extern "C" void kernel_launch(void* const* d_in, const int* in_sizes, int n_in,
                              void* d_out, int out_size, void* d_ws, size_t ws_size,
                              hipStream_t stream) {
    (void)n_in; (void)out_size; (void)d_ws; (void)ws_size;

    const int*   indices = (const int*)  d_in[0];
    const float* Wm_pos  = (const float*)d_in[1];
    const float* Wv_pos  = (const float*)d_in[2];
    const float* Wm_het  = (const float*)d_in[3];
    const float* Wv_het  = (const float*)d_in[4];

    const int B  = in_sizes[0];
    const int N  = in_sizes[3];                 // Wm_het has N elements
    const int K  = (N > 0) ? (in_sizes[1] / N) : 1;
    const int KC = K / 4;                       // float4 chunks per row

    float* out = (float*)d_out;
    float* pm = out;                                     // [B, K]
    float* pv = out + (size_t)B * (size_t)K;             // [B, K]
    float* hm = out + (size_t)2 * (size_t)B * (size_t)K; // [B]
    float* hv = hm + B;                                  // [B]

    // Persistent grid: 4096 blocks x 8 wave32 = 32768 waves; ~32MB of b128
    // loads in flight — enough to saturate 23.3 TB/s on random rows.
    int blocks = 4096;
    const int maxWavesNeeded = (B + 7) / 8;
    if (blocks > maxWavesNeeded) blocks = maxWavesNeeded > 0 ? maxWavesNeeded : 1;
    const int threads = 256;
    const int nwaves = blocks * (threads / 32);

    if (KC == 32) {
        prior_gather_rows_k128<<<blocks, threads, 0, stream>>>(
            indices, (const v4f*)Wm_pos, (const v4f*)Wv_pos,
            (v4f*)pm, (v4f*)pv, B, nwaves);
    } else {
        prior_gather_rows_generic<<<blocks, threads, 0, stream>>>(
            indices, (const v4f*)Wm_pos, (const v4f*)Wv_pos,
            (v4f*)pm, (v4f*)pv, B, KC, nwaves);
    }

    const int hblocks = (B + 255) / 256;
    prior_gather_het<<<hblocks, 256, 0, stream>>>(
        indices, Wm_het, Wv_het, hm, hv, B);
}